// BatchNormSPDMean_19877108646143
// MI455X (gfx1250) — compile-verified
//
#include <hip/hip_runtime.h>
#include <math.h>
#include <stdint.h>

// ---------------------------------------------------------------------------
// SPD batch-norm (Karcher mean + whitening + congruence bias) on MI455X.
// GEMM-only matrix functions -> every FLOP is V_WMMA_F32_16X16X4_F32 (wave32).
// Bulk tile movement (global<->LDS) uses the CDNA5 async-to-LDS path
// (GLOBAL_LOAD_ASYNC_TO_LDS_B128 / GLOBAL_STORE_ASYNC_FROM_LDS_B128,
// ASYNCcnt-tracked, VGPR-bypassing).
// LDS matrices padded to stride 68 floats: 16B-aligned rows for async b128,
// 8B-aligned fragment pairs, lanes spread across the 64 LDS banks.
// ---------------------------------------------------------------------------

typedef float v2f __attribute__((ext_vector_type(2)));
typedef float v8f __attribute__((ext_vector_type(8)));

#define N      64
#define NN     4096            // 64*64 logical elements
#define LDN    68              // padded LDS row stride (272B = 16*17)
#define MSZ    (N * LDN)       // floats per padded LDS matrix (4352)
#define BATCH  8192
#define NPART  64              // partial accumulators for the log-mean
#define TPB    256             // 8 wave32s per block

// Dynamic LDS layout used by all matrix kernels:
//   b0 | b1 | b2 | Bt(transpose stage) | red[TPB]
#define SMEM_FLOATS (4 * MSZ + TPB)
#define SMEM_BYTES  (SMEM_FLOATS * sizeof(float))

__device__ __forceinline__ v8f wmma_f32(v2f a, v2f b, v8f c) {
  return __builtin_amdgcn_wmma_f32_16x16x4_f32(
      /*neg_a=*/false, a, /*neg_b=*/false, b,
      /*c_mod=*/(short)0, c, /*reuse_a=*/false, /*reuse_b=*/false);
}

// ---------------------------------------------------------------------------
// CDNA5 async global<->LDS helpers (ASYNCcnt-tracked, bypass VGPRs).
// Generic pointers to __shared__ have the LDS byte offset in their low 32
// bits (the shared aperture lives in addr[63:32] per the ISA aperture check).
// ---------------------------------------------------------------------------
__device__ __forceinline__ void async_wait0() {
  asm volatile("s_wait_asynccnt 0x0" ::: "memory");
}

// Copy dense 64x64 global matrix -> padded LDS matrix (16B chunks).
__device__ __forceinline__ void lds_load_mat_async(float* dst,
                                                   const float* __restrict__ src) {
  for (int c = threadIdx.x; c < NN / 4; c += TPB) {      // 1024 chunks, 4/thread
    const int row = c >> 4, col4 = (c & 15) << 2;        // 16 chunks per row
    const uint32_t loff = (uint32_t)(uintptr_t)(dst + row * LDN + col4);
    const float* g = src + row * N + col4;
    asm volatile("global_load_async_to_lds_b128 %0, %1, off"
                 :: "v"(loff), "v"(g) : "memory");
  }
}

// Copy padded LDS matrix -> dense 64x64 global matrix (16B chunks).
__device__ __forceinline__ void lds_store_mat_async(float* __restrict__ dst,
                                                    const float* src) {
  for (int c = threadIdx.x; c < NN / 4; c += TPB) {
    const int row = c >> 4, col4 = (c & 15) << 2;
    const uint32_t loff = (uint32_t)(uintptr_t)(src + row * LDN + col4);
    float* g = dst + row * N + col4;
    asm volatile("global_store_async_from_lds_b128 %0, %1, off"
                 :: "v"(g), "v"(loff) : "memory");
  }
}

// ---------------------------------------------------------------------------
// Workgroup-cooperative 64x64 GEMM on padded LDS matrices:
//   C = scale * (A x Bop) + diag * I,   Bop = B (TRANSB=false) or B^T (true)
// Stage 1: cooperatively write Bop^T into Bt (so B fragments are contiguous).
// Stage 2: 8 waves; wave w owns row band (w/2)*16 and column tiles
//          (w&1)*32, (w&1)*32+16 -> one A-fragment stream feeds 2 WMMAs per k.
// Aliasing-safe: all LDS reads finish before the mid-barrier, so C may alias
// A and/or B (accumulators live entirely in VGPRs).
// ---------------------------------------------------------------------------
template <bool TRANSB>
__device__ __forceinline__ void gemm64(const float* A, const float* B, float* C,
                                       float* Bt, float scale, float diag) {
  const int tid  = threadIdx.x;
  const int lane = tid & 31;
  const int wave = tid >> 5;             // 0..7
  const int half = lane >> 4;            // 0/1  (ISA lane-group)
  const int l16  = lane & 15;

  __syncthreads();                       // A/B producers visible
  // Stage Bop^T into Bt:  Bt[n*LDN + k] = Bop[k][n]
  for (int i = tid; i < NN; i += TPB) {
    const int r = i >> 6, c = i & 63;
    if (TRANSB) Bt[r * LDN + c] = B[r * LDN + c];   // Bop=B^T -> direct copy
    else        Bt[c * LDN + r] = B[r * LDN + c];   // Bop=B   -> transpose
  }
  __syncthreads();

  const int ti  = (wave >> 1) * 16;      // C row band
  const int tj0 = (wave & 1) * 32;       // first column tile
  const int tj1 = tj0 + 16;              // second column tile

  const float* Ab = A  + (ti  + l16) * LDN;   // per-lane base, imm offsets
  const float* B0 = Bt + (tj0 + l16) * LDN;
  const float* B1 = Bt + (tj1 + l16) * LDN;

  v8f c0 = {}, c1 = {};
#pragma unroll
  for (int k = 0; k < 16; ++k) {
    const int k0 = k * 4 + 2 * half;     // compile-time after unroll (+2*half)
    v2f a, b0, b1;
    a.x  = Ab[k0];  a.y  = Ab[k0 + 1];   // contiguous pair loads
    b0.x = B0[k0];  b0.y = B0[k0 + 1];
    b1.x = B1[k0];  b1.y = B1[k0 + 1];
    c0 = wmma_f32(a, b0, c0);
    c1 = wmma_f32(a, b1, c1);
  }

  __syncthreads();                       // all A/B/Bt reads done -> C may alias
#pragma unroll
  for (int r = 0; r < 8; ++r) {
    const int row = ti + r + 8 * half;   // C layout: vgpr r -> M=r / M=r+8
    float v0 = c0[r] * scale;
    float v1 = c1[r] * scale;
    const int col0 = tj0 + l16;
    const int col1 = tj1 + l16;
    if (row == col0) v0 += diag;
    if (row == col1) v1 += diag;
    C[row * LDN + col0] = v0;
    C[row * LDN + col1] = v1;
  }
  __syncthreads();                       // C visible to consumers
}

// ---------------------------------------------------------------------------
// Elementwise helpers on padded LDS matrices
// ---------------------------------------------------------------------------
__device__ __forceinline__ void lds_load_mat(float* dst, const float* __restrict__ src) {
  for (int i = threadIdx.x; i < NN; i += TPB)
    dst[(i >> 6) * LDN + (i & 63)] = src[i];
}
__device__ __forceinline__ void lds_store_mat(float* __restrict__ dst, const float* src) {
  for (int i = threadIdx.x; i < NN; i += TPB)
    dst[i] = src[(i >> 6) * LDN + (i & 63)];
}
__device__ __forceinline__ void lds_set_identity(float* M, float d) {
  for (int i = threadIdx.x; i < NN; i += TPB) {
    const int r = i >> 6, c = i & 63;
    M[r * LDN + c] = (r == c) ? d : 0.0f;
  }
}

// Block-wide Frobenius norm of a padded LDS matrix.
__device__ __forceinline__ float block_frob(const float* Mat, float* red) {
  float s = 0.0f;
  for (int i = threadIdx.x; i < NN; i += TPB) {
    const float v = Mat[(i >> 6) * LDN + (i & 63)];
    s += v * v;
  }
  red[threadIdx.x] = s;
  __syncthreads();
  for (int off = TPB / 2; off > 0; off >>= 1) {
    if (threadIdx.x < (unsigned)off) red[threadIdx.x] += red[threadIdx.x + off];
    __syncthreads();
  }
  const float r = sqrtf(red[0]);
  __syncthreads();
  return r;
}

// Coupled Newton-Schulz: Y(in) spectrum in (0,1]; Y -> Y^(1/2), Z -> Y^(-1/2).
__device__ __forceinline__ void ns_sqrt(float* Y, float* Z, float* T, float* Bt,
                                        int iters) {
  lds_set_identity(Z, 1.0f);
  for (int it = 0; it < iters; ++it) {
    gemm64<false>(Z, Y, T, Bt, -0.5f, 1.5f);   // T = 1.5 I - 0.5 Z Y
    gemm64<false>(Y, T, Y, Bt,  1.0f, 0.0f);   // Y = Y T   (alias-safe)
    gemm64<false>(T, Z, Z, Bt,  1.0f, 0.0f);   // Z = T Z   (alias-safe)
  }
}

// ---------------------------------------------------------------------------
// Kernels
// ---------------------------------------------------------------------------

// M0 = mean over batch (one HBM stream; 134 MB then resident in 192 MB L2).
__global__ void spd_mean_kernel(const float* __restrict__ X, float* __restrict__ M) {
  const int idx = blockIdx.x * blockDim.x + threadIdx.x;
  if (idx >= NN) return;
  float s = 0.0f;
  for (int b = 0; b < BATCH; ++b) s += X[(size_t)b * NN + idx];
  M[idx] = s * (1.0f / (float)BATCH);
}

__global__ void spd_zero_kernel(float* __restrict__ p, int n) {
  const int i = blockIdx.x * blockDim.x + threadIdx.x;
  if (i < n) p[i] = 0.0f;
}

// Single workgroup: Ms = sqrtm(M), Mis = invsqrtm(M) via scaled Newton-Schulz.
__global__ void spd_sqrt_invsqrt_kernel(const float* __restrict__ Mg,
                                        float* __restrict__ Msg,
                                        float* __restrict__ Misg) {
  extern __shared__ float smem[];
  float* b0  = smem;
  float* b1  = smem + MSZ;
  float* b2  = smem + 2 * MSZ;
  float* Bt  = smem + 3 * MSZ;
  float* red = smem + 4 * MSZ;

  lds_load_mat(b0, Mg);
  __syncthreads();
  const float c  = block_frob(b0, red);    // ||M||_F: spectrum of M/c in (0,1]
  const float rc = 1.0f / c;
  for (int i = threadIdx.x; i < NN; i += TPB)
    b0[(i >> 6) * LDN + (i & 63)] *= rc;
  ns_sqrt(b0, b1, b2, Bt, 12);
  const float sc  = sqrtf(c);
  const float isc = 1.0f / sc;
  for (int i = threadIdx.x; i < NN; i += TPB) {
    const int j = (i >> 6) * LDN + (i & 63);
    Msg[i]  = b0[j] * sc;                  // sqrtm(M)    = sqrt(c) * Y
    Misg[i] = b1[j] * isc;                 // invsqrtm(M) = Z / sqrt(c)
  }
}

// One workgroup per batch matrix:
//   Y = Mis X Mis;  L = logm(Y) by 3 NS square roots + 8-term log(I+E) Horner;
//   atomically accumulate L into one of NPART partial sums (L2-side atomics).
__global__ void spd_log_accum_kernel(const float* __restrict__ X,
                                     const float* __restrict__ Misg,
                                     float* __restrict__ partials) {
  extern __shared__ float smem[];
  float* b0  = smem;
  float* b1  = smem + MSZ;
  float* b2  = smem + 2 * MSZ;
  float* Bt  = smem + 3 * MSZ;
  float* red = smem + 4 * MSZ;

  const size_t base = (size_t)blockIdx.x * NN;
  lds_load_mat_async(b0, X + base);            // async global -> LDS, no VGPRs
  lds_load_mat_async(b1, Misg);
  async_wait0();                               // ASYNCcnt == 0 for this wave
  gemm64<false>(b1, b0, b2, Bt, 1.0f, 0.0f);   // tmp = Mis X (entry barrier)
  gemm64<false>(b2, b1, b0, Bt, 1.0f, 0.0f);   // Y   = tmp Mis

  const float c  = block_frob(b0, red);        // scale: Y/c spectrum in (0,1]
  const float rc = 1.0f / c;
  for (int i = threadIdx.x; i < NN; i += TPB)
    b0[(i >> 6) * LDN + (i & 63)] *= rc;

  for (int s = 0; s < 3; ++s) ns_sqrt(b0, b1, b2, Bt, 6);  // R = (Y/c)^(1/8)

  for (int i = threadIdx.x; i < NN; i += TPB) {             // E = R - I
    const int r = i >> 6, cc = i & 63;
    b1[r * LDN + cc] = b0[r * LDN + cc] - ((r == cc) ? 1.0f : 0.0f);
  }
  lds_set_identity(b2, 0.125f);                             // S = (1/8) I
  for (int k = 7; k >= 1; --k)
    gemm64<false>(b1, b2, b2, Bt, -1.0f, 1.0f / (float)k);  // S = (1/k)I - E S
  const float logc = logf(c);
  gemm64<false>(b1, b2, b0, Bt, 8.0f, logc);   // L = 2^3 * E S + log(c) I

  float* part = partials + (size_t)(blockIdx.x & (NPART - 1)) * NN;
  for (int i = threadIdx.x; i < NN; i += TPB)
    __hip_atomic_fetch_add(&part[i], b0[(i >> 6) * LDN + (i & 63)],
                           __ATOMIC_RELAXED, __HIP_MEMORY_SCOPE_AGENT);
}

// Single workgroup: L = mean of logs; M <- sym(Ms expm(L) Ms).
__global__ void spd_karcher_update_kernel(const float* __restrict__ partials,
                                          const float* __restrict__ Msg,
                                          float* __restrict__ Mg) {
  extern __shared__ float smem[];
  float* b0 = smem;
  float* b1 = smem + MSZ;
  float* b2 = smem + 2 * MSZ;
  float* Bt = smem + 3 * MSZ;

  for (int i = threadIdx.x; i < NN; i += TPB) {
    float s = 0.0f;
    for (int p = 0; p < NPART; ++p) s += partials[(size_t)p * NN + i];
    b0[(i >> 6) * LDN + (i & 63)] =
        s * (1.0f / (float)BATCH) * (1.0f / 64.0f);         // L' = L / 2^6
  }
  lds_set_identity(b1, 1.0f);                               // S = I
  for (int k = 8; k >= 1; --k)
    gemm64<false>(b0, b1, b1, Bt, 1.0f / (float)k, 1.0f);   // S = I + (1/k)L'S
  for (int j = 0; j < 6; ++j)
    gemm64<false>(b1, b1, b1, Bt, 1.0f, 0.0f);              // S = S S
  lds_load_mat(b0, Msg);
  gemm64<false>(b0, b1, b2, Bt, 1.0f, 0.0f);                // t  = Ms expm(L)
  gemm64<false>(b2, b0, b1, Bt, 1.0f, 0.0f);                // M2 = t Ms
  for (int i = threadIdx.x; i < NN; i += TPB) {
    const int r = i >> 6, c = i & 63;
    Mg[i] = 0.5f * (b1[r * LDN + c] + b1[c * LDN + r]);     // sym
  }
}

// Single workgroup: Bs = expm(0.5 sym(bias_raw)); P = Bs @ W (W = invsqrtm(M)).
__global__ void spd_bias_prep_kernel(const float* __restrict__ bias_raw,
                                     const float* __restrict__ Wg,
                                     float* __restrict__ Pg) {
  extern __shared__ float smem[];
  float* b0 = smem;
  float* b1 = smem + MSZ;
  float* b2 = smem + 2 * MSZ;
  float* Bt = smem + 3 * MSZ;

  for (int i = threadIdx.x; i < NN; i += TPB) {
    const int r = i >> 6, c = i & 63;
    // A = 0.5*sym(bias_raw) = 0.25*(b + b^T); pre-divide by 2^4 for squaring
    b0[r * LDN + c] =
        0.25f * (bias_raw[i] + bias_raw[c * N + r]) * (1.0f / 16.0f);
  }
  lds_set_identity(b1, 1.0f);
  for (int k = 8; k >= 1; --k)
    gemm64<false>(b0, b1, b1, Bt, 1.0f / (float)k, 1.0f);   // Taylor Horner
  for (int j = 0; j < 4; ++j)
    gemm64<false>(b1, b1, b1, Bt, 1.0f, 0.0f);              // square back
  lds_load_mat(b0, Wg);
  gemm64<false>(b1, b0, b2, Bt, 1.0f, 0.0f);                // P = Bs W
  lds_store_mat(Pg, b2);
}

// One workgroup per batch matrix: out = P X P^T  (= Bs W X W Bs).
__global__ void spd_apply_kernel(const float* __restrict__ X,
                                 const float* __restrict__ Pg,
                                 float* __restrict__ out) {
  extern __shared__ float smem[];
  float* b0 = smem;
  float* b1 = smem + MSZ;
  float* b2 = smem + 2 * MSZ;
  float* Bt = smem + 3 * MSZ;

  const size_t base = (size_t)blockIdx.x * NN;
  lds_load_mat_async(b0, X + base);            // async global -> LDS
  lds_load_mat_async(b1, Pg);
  async_wait0();
  gemm64<false>(b1, b0, b2, Bt, 1.0f, 0.0f);   // t   = P X
  gemm64<true >(b2, b1, b0, Bt, 1.0f, 0.0f);   // out = t P^T
  lds_store_mat_async(out + base, b0);         // async LDS -> global
  async_wait0();                               // drain before wave end
}

// ---------------------------------------------------------------------------
extern "C" void kernel_launch(void* const* d_in, const int* in_sizes, int n_in,
                              void* d_out, int out_size, void* d_ws, size_t ws_size,
                              hipStream_t stream) {
  (void)in_sizes; (void)n_in; (void)out_size; (void)ws_size;
  const float* data     = (const float*)d_in[0];   // (8192, 64, 64) f32
  const float* bias_raw = (const float*)d_in[1];   // (64, 64) f32
  float* outp = (float*)d_out;

  // Workspace layout (floats): M | Ms | Mis | P | partials[NPART*NN]
  float* M    = (float*)d_ws;
  float* Ms   = M   + NN;
  float* Mis  = Ms  + NN;
  float* P    = Mis + NN;
  float* part = P   + NN;
  const int npart_elems = NPART * NN;

  spd_mean_kernel<<<(NN + TPB - 1) / TPB, TPB, 0, stream>>>(data, M);

  for (int it = 0; it < 5; ++it) {
    spd_sqrt_invsqrt_kernel<<<1, TPB, SMEM_BYTES, stream>>>(M, Ms, Mis);
    spd_zero_kernel<<<(npart_elems + TPB - 1) / TPB, TPB, 0, stream>>>(part, npart_elems);
    spd_log_accum_kernel<<<BATCH, TPB, SMEM_BYTES, stream>>>(data, Mis, part);
    spd_karcher_update_kernel<<<1, TPB, SMEM_BYTES, stream>>>(part, Ms, M);
  }

  spd_sqrt_invsqrt_kernel<<<1, TPB, SMEM_BYTES, stream>>>(M, Ms, Mis);  // Mis = W
  spd_bias_prep_kernel<<<1, TPB, SMEM_BYTES, stream>>>(bias_raw, Mis, P);
  spd_apply_kernel<<<BATCH, TPB, SMEM_BYTES, stream>>>(data, P, outp);
}